// MatrixVectorScaledDotProductAttention_377957122129
// MI455X (gfx1250) — compile-verified
//
#include <hip/hip_runtime.h>
#include <hip/hip_bf16.h>

typedef __attribute__((ext_vector_type(2))) float v2f;
typedef __attribute__((ext_vector_type(8))) float v8f;

#define NB_   2048
#define L_    2048
#define DK_   64
#define DV_   64
#define INV_TEMP (0.125f)   // 1 / 8.0

// One workgroup per batch row b. 256 threads = 8 waves (wave32).
// Phase 1: scores[l] = (K[b,l,:] . q[b,:]) / T  via V_WMMA_F32_16X16X4_F32
//          (q broadcast across the 16 B-columns; D rows carry 16 scores).
// Phase 2: block softmax in LDS; write attn (prob) to global + LDS.
// Phase 3: out[b,e] = sum_l p[l] * V[b,l,e]  (coalesced V streaming, fmaf).
__launch_bounds__(256, 2)
__global__ void mv_sdpa_kernel(const float* __restrict__ q,
                               const float* __restrict__ k,
                               const float* __restrict__ v,
                               float* __restrict__ out,    // [NB, DV]
                               float* __restrict__ attn)   // [NB, L]
{
    __shared__ float q_s[DK_];
    __shared__ float p_s[L_];          // scores, then probabilities
    __shared__ float red_s[256];       // reduction scratch
    __shared__ float acc_s[4 * DV_];   // PV partial sums

    const int b    = blockIdx.x;
    const int tid  = threadIdx.x;
    const int wave = tid >> 5;         // 0..7
    const int lane = tid & 31;         // wave32

    if (tid < DK_) q_s[tid] = q[b * DK_ + tid];
    __syncthreads();

    // ---------------- Phase 1: QK^T scores via WMMA f32 16x16x4 ----------------
    // A 16x4 f32 layout (2 VGPRs/lane): lanes 0-15 hold rows 0-15 at K=kpair{0,1},
    // lanes 16-31 hold rows 0-15 at K={2,3}. B 4x16 mirrors the K striping, with
    // N across lanes; broadcasting q over all N makes every D column the score.
    const float* kb   = k + (size_t)b * L_ * DK_;
    const int    arow  = lane & 15;
    const int    kpair = (lane >> 4) * 2;   // 0 or 2

    // each wave covers 256 consecutive l-values: 16 tiles of 16 rows
    for (int t = 0; t < 16; ++t) {
        const int l0 = wave * 256 + t * 16;
        v8f c = {};   // accumulator: 16x16 f32 D tile
        const float* arow_ptr = kb + (size_t)(l0 + arow) * DK_ + kpair;
        #pragma unroll
        for (int d0 = 0; d0 < DK_; d0 += 4) {
            v2f a = *(const v2f*)(arow_ptr + d0);          // K streams into A
            v2f bm;
            bm.x = q_s[d0 + kpair];
            bm.y = q_s[d0 + kpair + 1];
            // 8 args: (neg_a, A, neg_b, B, c_mod, C, reuse_a, reuse_b)
            c = __builtin_amdgcn_wmma_f32_16x16x4_f32(
                    false, a, false, bm, (short)0, c, false, false);
        }
        // D[i,j] identical over j: lane 0 holds rows 0-7, lane 16 rows 8-15
        if ((lane & 15) == 0) {
            const int base = l0 + (lane >> 4) * 8;
            #pragma unroll
            for (int r = 0; r < 8; ++r)
                p_s[base + r] = c[r] * INV_TEMP;
        }
    }
    __syncthreads();

    // ---------------- Phase 2: softmax over l (block-wide) ----------------
    // each thread owns 8 consecutive l-values
    const int lbase = tid * 8;
    float m = -INFINITY;
    #pragma unroll
    for (int i = 0; i < 8; ++i) m = fmaxf(m, p_s[lbase + i]);
    red_s[tid] = m;
    __syncthreads();
    for (int s = 128; s > 0; s >>= 1) {
        if (tid < s) red_s[tid] = fmaxf(red_s[tid], red_s[tid + s]);
        __syncthreads();
    }
    const float gmax = red_s[0];
    __syncthreads();

    float e[8];
    float lsum = 0.0f;
    #pragma unroll
    for (int i = 0; i < 8; ++i) {
        e[i] = __expf(p_s[lbase + i] - gmax);
        lsum += e[i];
    }
    red_s[tid] = lsum;
    __syncthreads();
    for (int s = 128; s > 0; s >>= 1) {
        if (tid < s) red_s[tid] += red_s[tid + s];
        __syncthreads();
    }
    const float inv_sum = 1.0f / red_s[0];
    __syncthreads();

    // write probabilities to LDS and to the attn output (float4 x2 per thread)
    float4 p01, p23;
    p01.x = e[0] * inv_sum; p01.y = e[1] * inv_sum;
    p01.z = e[2] * inv_sum; p01.w = e[3] * inv_sum;
    p23.x = e[4] * inv_sum; p23.y = e[5] * inv_sum;
    p23.z = e[6] * inv_sum; p23.w = e[7] * inv_sum;
    p_s[lbase + 0] = p01.x; p_s[lbase + 1] = p01.y;
    p_s[lbase + 2] = p01.z; p_s[lbase + 3] = p01.w;
    p_s[lbase + 4] = p23.x; p_s[lbase + 5] = p23.y;
    p_s[lbase + 6] = p23.z; p_s[lbase + 7] = p23.w;
    float4* attn_v4 = (float4*)(attn + (size_t)b * L_ + lbase);
    attn_v4[0] = p01;
    attn_v4[1] = p23;
    __syncthreads();

    // ---------------- Phase 3: out[b,e] = sum_l p[l] * V[b,l,e] ----------------
    // thread -> (e = tid%64, chunk = tid/64 of 512 l's); wave reads 128B/row of V
    const float* vb    = v + (size_t)b * L_ * DV_;
    const int    eidx  = tid & 63;
    const int    chunk = tid >> 6;          // 0..3
    const int    lbeg  = chunk * 512;
    float acc = 0.0f;
    #pragma unroll 4
    for (int l = lbeg; l < lbeg + 512; ++l)
        acc = fmaf(p_s[l], vb[(size_t)l * DV_ + eidx], acc);
    acc_s[chunk * DV_ + eidx] = acc;
    __syncthreads();
    if (tid < DV_) {
        out[b * DV_ + tid] = acc_s[tid] + acc_s[DV_ + tid] +
                             acc_s[2 * DV_ + tid] + acc_s[3 * DV_ + tid];
    }
}

extern "C" void kernel_launch(void* const* d_in, const int* in_sizes, int n_in,
                              void* d_out, int out_size, void* d_ws, size_t ws_size,
                              hipStream_t stream) {
    (void)in_sizes; (void)n_in; (void)d_ws; (void)ws_size; (void)out_size;
    const float* q = (const float*)d_in[0];   // [NB, DK]
    const float* k = (const float*)d_in[1];   // [NB, L, DK]
    const float* v = (const float*)d_in[2];   // [NB, L, DV]
    float* out  = (float*)d_out;                       // [NB, DV] first
    float* attn = (float*)d_out + (size_t)NB_ * DV_;   // then [NB, L]

    mv_sdpa_kernel<<<NB_, 256, 0, stream>>>(q, k, v, out, attn);
}